// TemporalLightGCNLayer_66967130079600
// MI455X (gfx1250) — compile-verified
//
#include <hip/hip_runtime.h>
#include <stdint.h>

// Temporal LightGCN layer: h_new[dst] += h[src] * (norm * exp(-lam*dt))
// Memory-bound scatter-gather: ~0.69 GB moved per call -> ~30us floor @ 23.3 TB/s.
// CDNA5 paths used: global_load_async_to_lds_b64/b32 (ASYNCcnt) for edge-metadata
// staging, s_wait_asynccnt, native fp32 global atomics for the scatter.

#define EDGES_PER_BLOCK 16
#define BLOCK_THREADS   256   // 8 wave32 waves; 16 lanes cooperate per edge

__global__ __launch_bounds__(BLOCK_THREADS)
void temporal_gcn_scatter(const float* __restrict__ h,
                          const long long* __restrict__ src,
                          const long long* __restrict__ dst,
                          const float* __restrict__ dt,
                          const float* __restrict__ nrm,
                          const float* __restrict__ lam_p,
                          float* __restrict__ out,
                          int E)
{
    // Per-block staged edge metadata (384 bytes of LDS).
    __shared__ long long s_src[EDGES_PER_BLOCK];
    __shared__ long long s_dst[EDGES_PER_BLOCK];
    __shared__ float     s_dt [EDGES_PER_BLOCK];
    __shared__ float     s_nrm[EDGES_PER_BLOCK];

    const int tid = threadIdx.x;
    const int e0  = blockIdx.x * EDGES_PER_BLOCK;

    // ---- Wave 0 stages all edge metadata via async tensor-path copies ----
    // lanes 0..15: src[e0+l] (b64) and dt[e0+l] (b32)
    // lanes 16..31: dst[e0+l-16] (b64) and nrm[e0+l-16] (b32)
    if (tid < 32) {
        const int  lane  = tid;
        const bool hiHalf = lane >= 16;
        const int  el    = lane & 15;
        if (e0 + el < E) {
            const long long* gp64 = hiHalf ? (dst + e0 + el) : (src + e0 + el);
            uint64_t ga64 = (uint64_t)(uintptr_t)gp64;
            uint32_t la64 = (uint32_t)(uintptr_t)(hiHalf ? (void*)&s_dst[el]
                                                         : (void*)&s_src[el]);
            // ASYNCcnt-tracked DMA: copies 8 bytes per lane from the global address in ga64 to the LDS address in la64
            asm volatile("global_load_async_to_lds_b64 %0, %1, off"
                         :: "v"(la64), "v"(ga64) : "memory");

            const float* gp32 = hiHalf ? (nrm + e0 + el) : (dt + e0 + el);
            uint64_t ga32 = (uint64_t)(uintptr_t)gp32;
            uint32_t la32 = (uint32_t)(uintptr_t)(hiHalf ? (void*)&s_nrm[el]
                                                         : (void*)&s_dt[el]);
            asm volatile("global_load_async_to_lds_b32 %0, %1, off"
                         :: "v"(la32), "v"(ga32) : "memory");
        }
        // Wait for the async copies to land in LDS (issuing wave only).
        asm volatile("s_wait_asynccnt 0x0" ::: "memory");
    }
    __syncthreads();  // publish LDS to the other 7 waves

    // ---- Compute + scatter ----
    const float lam = fmaxf(lam_p[0], 0.0f) + 1.0e-4f;

    const int el   = tid >> 4;    // edge slot within block: 0..15
    const int part = tid & 15;    // which float4 of the 64-float row
    const int e    = e0 + el;
    if (e < E) {
        const int   s = (int)s_src[el];      // indices < 100000 fit in 32 bits
        const int   d = (int)s_dst[el];
        const float w = s_nrm[el] * __expf(-lam * s_dt[el]);

        // 16B coalesced gather of this thread's slice of h[src]
        const float4 v = *(const float4*)(h + (size_t)s * 64 + part * 4);

        float* o = out + (size_t)d * 64 + part * 4;
        // Relaxed agent-scope fp32 adds -> native global_atomic_add_f32
        __hip_atomic_fetch_add(o + 0, v.x * w, __ATOMIC_RELAXED, __HIP_MEMORY_SCOPE_AGENT);
        __hip_atomic_fetch_add(o + 1, v.y * w, __ATOMIC_RELAXED, __HIP_MEMORY_SCOPE_AGENT);
        __hip_atomic_fetch_add(o + 2, v.z * w, __ATOMIC_RELAXED, __HIP_MEMORY_SCOPE_AGENT);
        __hip_atomic_fetch_add(o + 3, v.w * w, __ATOMIC_RELAXED, __HIP_MEMORY_SCOPE_AGENT);
    }
}

extern "C" void kernel_launch(void* const* d_in, const int* in_sizes, int n_in,
                              void* d_out, int out_size, void* d_ws, size_t ws_size,
                              hipStream_t stream) {
    (void)n_in; (void)d_ws; (void)ws_size;

    const float*     h    = (const float*)d_in[0];
    const long long* src  = (const long long*)d_in[1];
    const long long* dstI = (const long long*)d_in[2];
    const float*     dt   = (const float*)d_in[3];
    const float*     nrm  = (const float*)d_in[4];
    const float*     lam  = (const float*)d_in[5];
    float*           out  = (float*)d_out;

    const int E = in_sizes[1];  // number of edges (src array length)

    // Zero the accumulator every call (graph-capture-safe, deterministic).
    hipMemsetAsync(d_out, 0, (size_t)out_size * sizeof(float), stream);

    const int blocks = (E + EDGES_PER_BLOCK - 1) / EDGES_PER_BLOCK;
    temporal_gcn_scatter<<<blocks, BLOCK_THREADS, 0, stream>>>(
        h, src, dstI, dt, nrm, lam, out, E);
}